// SSIMLoss_63617055588533
// MI455X (gfx1250) — compile-verified
//
#include <hip/hip_runtime.h>

// ---------------- problem constants ----------------
#define IMG_H 512
#define IMG_W 512
#define NBATCH 16
#define NCHAN 3
#define WSZ 11          // gaussian window size
#define PAD 5           // WSZ/2
#define TILE 32         // block output tile (32x32), 4 waves of 16x16
#define RAWDIM 42       // TILE + 2*PAD
#define RAWSTR 44       // padded stride (cols 42,43 zero-filled)
#define HROWS 26        // 16 + 2*PAD intermediate rows per wave tile
#define HSTR 17         // padded stride for H staging
#define NFIELD 5        // img1, img2, img1^2, img2^2, img1*img2
#define KCHUNKS 7       // ceil(26/4) K-chunks of the banded GEMM

typedef float v2f __attribute__((ext_vector_type(2)));
typedef float v8f __attribute__((ext_vector_type(8)));

__device__ __forceinline__ int imin(int a, int b) { return a < b ? a : b; }

// band weight: value of unnormalized gaussian g[d] = exp(-(d-5)^2/(2*1.5^2))
// for tap distance d in [0,11), else 0.
__device__ __forceinline__ float band_val(int d) {
    float fd = (float)(d - PAD);
    float g = __expf(-(fd * fd) * (1.0f / 4.5f));
    return (d >= 0 && d < WSZ) ? g : 0.0f;
}

__global__ __launch_bounds__(128) void ssim_tile_kernel(
    const float* __restrict__ img1,
    const float* __restrict__ img2,
    float* __restrict__ partial)
{
    __shared__ float raw1[RAWDIM * RAWSTR];
    __shared__ float raw2[RAWDIM * RAWSTR];
    __shared__ float hbuf[NFIELD * 4 * HROWS * HSTR]; // per field, per wave
    __shared__ float red[128];

    const int tid    = threadIdx.x;
    const int lane   = tid & 31;
    const int wave   = tid >> 5;
    const int laneHi = lane >> 4;   // 0 for lanes 0-15, 1 for 16-31
    const int laneLo = lane & 15;

    // ---- decompose block index: (batch*channel, tile_y, tile_x) ----
    const int tiles_x = IMG_W / TILE;               // 16
    const int tiles_per_ch = tiles_x * (IMG_H / TILE); // 256
    int bidx = blockIdx.x;
    int bc  = bidx / tiles_per_ch;                  // 0..47
    int t   = bidx % tiles_per_ch;
    int by0 = (t / tiles_x) * TILE;
    int bx0 = (t % tiles_x) * TILE;

    const float* base1 = img1 + (size_t)bc * IMG_H * IMG_W;
    const float* base2 = img2 + (size_t)bc * IMG_H * IMG_W;

    // ---- cooperative load of zero-padded halo tiles into LDS ----
    for (int e = tid; e < RAWDIM * RAWSTR; e += 128) {
        int r = e / RAWSTR, c = e % RAWSTR;
        int gy = by0 - PAD + r;
        int gx = bx0 - PAD + c;
        bool ok = (c < RAWDIM) & (gy >= 0) & (gy < IMG_H) & (gx >= 0) & (gx < IMG_W);
        size_t off = (size_t)gy * IMG_W + gx;
        raw1[e] = ok ? base1[off] : 0.0f;
        raw2[e] = ok ? base2[off] : 0.0f;
    }
    __syncthreads();

    // ---- per-lane banded gaussian weights, shared by both passes ----
    // Layout formula (A 16x4 f32 and B 4x16 f32): this lane's slot s holds
    // K-index k = 2*laneHi + s; band row j = 4*q + k; band col/row p = laneLo.
    v2f bW[KCHUNKS];
#pragma unroll
    for (int q = 0; q < KCHUNKS; ++q) {
#pragma unroll
        for (int s = 0; s < 2; ++s) {
            int j = 4 * q + 2 * laneHi + s;
            bW[q][s] = band_val(j - laneLo);
        }
    }

    // wave's 16x16 sub-tile origin inside the 32x32 block tile
    const int ty0 = (wave >> 1) * 16;
    const int tx0 = (wave & 1) * 16;

    // ================= horizontal pass (2 row-groups x 7 chunks x 5 fields) ==
#pragma unroll
    for (int grp = 0; grp < 2; ++grp) {
        const int jb = grp * 10;              // group rows cover H rows jb..jb+15
        const int rrow = ty0 + jb + laneLo;   // raw row of this lane's A rows
        v8f a0 = {}, a1 = {}, a2 = {}, a3 = {}, a4 = {};
#pragma unroll
        for (int q = 0; q < KCHUNKS; ++q) {
            int rc = tx0 + 4 * q + 2 * laneHi; // raw col for slot 0 (slot 1 = +1)
            const float* r1p = &raw1[rrow * RAWSTR + rc];
            const float* r2p = &raw2[rrow * RAWSTR + rc];
            float x0 = r1p[0], x1 = r1p[1];
            float y0 = r2p[0], y1 = r2p[1];
            v2f A0 = {x0, x1};
            v2f A1 = {y0, y1};
            v2f A2 = {x0 * x0, x1 * x1};
            v2f A3 = {y0 * y0, y1 * y1};
            v2f A4 = {x0 * y0, x1 * y1};
            v2f Bq = bW[q];
            a0 = __builtin_amdgcn_wmma_f32_16x16x4_f32(false, A0, false, Bq, (short)0, a0, false, false);
            a1 = __builtin_amdgcn_wmma_f32_16x16x4_f32(false, A1, false, Bq, (short)0, a1, false, false);
            a2 = __builtin_amdgcn_wmma_f32_16x16x4_f32(false, A2, false, Bq, (short)0, a2, false, false);
            a3 = __builtin_amdgcn_wmma_f32_16x16x4_f32(false, A3, false, Bq, (short)0, a3, false, false);
            a4 = __builtin_amdgcn_wmma_f32_16x16x4_f32(false, A4, false, Bq, (short)0, a4, false, false);
        }
        // store D (C/D layout: row = jb + r + 8*laneHi, col = laneLo)
        int drow = jb + 8 * laneHi;
#pragma unroll
        for (int r = 0; r < 8; ++r) {
            int off = (drow + r) * HSTR + laneLo;
            hbuf[(0 * 4 + wave) * HROWS * HSTR + off] = a0[r];
            hbuf[(1 * 4 + wave) * HROWS * HSTR + off] = a1[r];
            hbuf[(2 * 4 + wave) * HROWS * HSTR + off] = a2[r];
            hbuf[(3 * 4 + wave) * HROWS * HSTR + off] = a3[r];
            hbuf[(4 * 4 + wave) * HROWS * HSTR + off] = a4[r];
        }
    }
    __syncthreads();

    // ================= vertical pass (7 chunks x 5 fields) ===================
    v8f vacc[NFIELD];
#pragma unroll
    for (int f = 0; f < NFIELD; ++f) {
        const float* hb = &hbuf[(f * 4 + wave) * HROWS * HSTR];
        v8f acc = {};
#pragma unroll
        for (int q = 0; q < KCHUNKS; ++q) {
            int j0 = 4 * q + 2 * laneHi;
            // rows 26,27 in the last chunk carry zero band weight; clamp the read
            v2f Bv;
            Bv[0] = hb[imin(j0, HROWS - 1) * HSTR + laneLo];
            Bv[1] = hb[imin(j0 + 1, HROWS - 1) * HSTR + laneLo];
            acc = __builtin_amdgcn_wmma_f32_16x16x4_f32(false, bW[q], false, Bv, (short)0, acc, false, false);
        }
        vacc[f] = acc;
    }

    // ================= SSIM map + block reduction ============================
    const float C1 = 0.0001f;  // 0.01^2
    const float C2 = 0.0009f;  // 0.03^2
    float lsum = 0.0f;
#pragma unroll
    for (int r = 0; r < 8; ++r) {
        float mu1 = vacc[0][r], mu2 = vacc[1][r];
        float x11 = vacc[2][r], x22 = vacc[3][r], x12 = vacc[4][r];
        float mu1s = mu1 * mu1, mu2s = mu2 * mu2, mu12 = mu1 * mu2;
        float s1  = x11 - mu1s;
        float s2  = x22 - mu2s;
        float s12 = x12 - mu12;
        float num = (2.0f * mu12 + C1) * (2.0f * s12 + C2);
        float den = (mu1s + mu2s + C1) * (s1 + s2 + C2);
        lsum += num / den;
    }
    red[tid] = lsum;
    __syncthreads();
#pragma unroll
    for (int s = 64; s > 0; s >>= 1) {
        if (tid < s) red[tid] += red[tid + s];
        __syncthreads();
    }
    if (tid == 0) partial[blockIdx.x] = red[0];
}

__global__ __launch_bounds__(256) void ssim_finalize(
    const float* __restrict__ partial, float* __restrict__ out,
    int nblocks, float invN)
{
    __shared__ float red[256];
    float s = 0.0f;
    for (int i = threadIdx.x; i < nblocks; i += 256) s += partial[i]; // fixed order
    red[threadIdx.x] = s;
    __syncthreads();
    for (int k = 128; k > 0; k >>= 1) {
        if (threadIdx.x < k) red[threadIdx.x] += red[threadIdx.x + k];
        __syncthreads();
    }
    if (threadIdx.x == 0) out[0] = 1.0f - red[0] * invN;
}

extern "C" void kernel_launch(void* const* d_in, const int* in_sizes, int n_in,
                              void* d_out, int out_size, void* d_ws, size_t ws_size,
                              hipStream_t stream) {
    const float* img1 = (const float*)d_in[0];
    const float* img2 = (const float*)d_in[1];
    float* out = (float*)d_out;
    float* partial = (float*)d_ws; // nblocks floats (~49 KB)

    const int nblocks = NBATCH * NCHAN * (IMG_H / TILE) * (IMG_W / TILE); // 12288
    ssim_tile_kernel<<<nblocks, 128, 0, stream>>>(img1, img2, partial);

    const float invN = 1.0f / (float)((size_t)NBATCH * NCHAN * IMG_H * IMG_W);
    ssim_finalize<<<1, 256, 0, stream>>>(partial, out, nblocks, invN);
}